// Attention_66125316489688
// MI455X (gfx1250) — compile-verified
//
#include <hip/hip_runtime.h>
#include <math.h>

typedef __attribute__((ext_vector_type(2))) float v2f;
typedef __attribute__((ext_vector_type(8))) float v8f;

#define NEGF (-3.402823466e38f)

static constexpr int    HN   = 12;
static constexpr int    HD   = 32;
static constexpr int    NTOK = 1024;
static constexpr int    MTOK = 1040;   // N + 16 memory tokens (= 65 * 16)
static constexpr int    NMEM = 16;
static constexpr int    KTOP = 64;
static constexpr int    LDS_STRIDE = 1044;            // 1040 padded (bank-conflict-free)
static constexpr size_t SQ = (size_t)8 * HN * NTOK * HD; // elements per q/k/v slab

__device__ __forceinline__ v8f wmma4(v2f a, v2f b, v8f c) {
  // D = A(16x4,f32) * B(4x16,f32) + C(16x16,f32)
  return __builtin_amdgcn_wmma_f32_16x16x4_f32(false, a, false, b, (short)0, c,
                                               false, false);
}

__device__ __forceinline__ float wred_sum(float v) {
#pragma unroll
  for (int o = 16; o > 0; o >>= 1) v += __shfl_xor(v, o, 32);
  return v;
}
__device__ __forceinline__ int wred_sumi(int v) {
#pragma unroll
  for (int o = 16; o > 0; o >>= 1) v += __shfl_xor(v, o, 32);
  return v;
}
__device__ __forceinline__ float wred_max(float v) {
#pragma unroll
  for (int o = 16; o > 0; o >>= 1) v = fmaxf(v, __shfl_xor(v, o, 32));
  return v;
}
// order-preserving float -> uint mapping
__device__ __forceinline__ unsigned f2key(float f) {
  unsigned u = __float_as_uint(f);
  return (u & 0x80000000u) ? ~u : (u | 0x80000000u);
}

// ---------------------------------------------------------------------------
// C[i][j] = sum_k A[i][k] * Bw[j][k]   (A: Mx K, Bw: N x K, both row-major)
// WG = 256 threads = 8 waves (2 row-tiles x 4 col-tiles); wave tile 16x16.
// ---------------------------------------------------------------------------
__global__ void __launch_bounds__(256) gemm_nt(const float* __restrict__ A,
                                               const float* __restrict__ Bw,
                                               float* __restrict__ C, int K,
                                               int Ncols) {
  const int lane = threadIdx.x & 31;
  const int wave = threadIdx.x >> 5;
  const int half = lane >> 4;   // 0: K pairs {0,1}; 1: K pairs {2,3}
  const int lr   = lane & 15;
  const int i0 = blockIdx.y * 32 + (wave >> 2) * 16;
  const int j0 = blockIdx.x * 64 + (wave & 3) * 16;
  const float* ar = A + (size_t)(i0 + lr) * K + 2 * half;
  const float* br = Bw + (size_t)(j0 + lr) * K + 2 * half;
  v8f acc = {};
#pragma unroll 8
  for (int k = 0; k < K; k += 4) {
    v2f a = *(const v2f*)(ar + k);
    v2f b = *(const v2f*)(br + k);
    acc = wmma4(a, b, acc);
  }
#pragma unroll
  for (int r = 0; r < 8; ++r)
    C[(size_t)(i0 + r + 8 * half) * Ncols + j0 + lr] = acc[r];
}

// ---------------------------------------------------------------------------
// Fused attention: grid = B*H*(N/64) blocks of 128 threads (4 waves).
// Each wave owns 16 query rows and a private 16 x 1040 score strip in LDS.
// ---------------------------------------------------------------------------
__global__ void __launch_bounds__(128) attn_kernel(
    const float* __restrict__ qkv, const float* __restrict__ scale,
    const float* __restrict__ memk, const float* __restrict__ memv,
    float* __restrict__ o2) {
  extern __shared__ float lds[];
  const int lane = threadIdx.x & 31;
  const int wave = threadIdx.x >> 5;
  const int half = lane >> 4;
  const int lr   = lane & 15;
  const int rblk = blockIdx.x & 15;  // 16 row-blocks of 64 per (b,h)
  const int bh   = blockIdx.x >> 4;
  const int h    = bh % HN;
  const int b    = bh / HN;
  const int n0   = rblk * 64 + wave * 16;
  float* sc = lds + (size_t)wave * 16 * LDS_STRIDE;
  const size_t qbase = ((size_t)(b * HN + h)) * NTOK * HD;

  // ---- Q fragments (row n0+lr; this lane-half's K pairs), l2-normalized ----
  const float* qrow = qkv + qbase + (size_t)(n0 + lr) * HD + 2 * half;
  v2f fq[8];
  float qn2 = 0.f;
#pragma unroll
  for (int kk = 0; kk < 8; ++kk) {
    v2f t = *(const v2f*)(qrow + 4 * kk);
    fq[kk] = t;
    qn2 += t.x * t.x + t.y * t.y;
  }
  qn2 += __shfl_xor(qn2, 16, 32);  // partner half of the same row
  const float qinv = 1.0f / fmaxf(sqrtf(qn2), 1e-12f);
#pragma unroll
  for (int kk = 0; kk < 8; ++kk) fq[kk] *= qinv;

  const float sch = scale[h];

  // ---- scores = (q_n . k_m) * scale, diagonal masked, into LDS ----
  for (int mt = 0; mt < 65; ++mt) {
    const float* krow =
        (mt < 64) ? qkv + SQ + qbase + (size_t)(mt * 16 + lr) * HD + 2 * half
                  : memk + ((size_t)h * NMEM + lr) * HD + 2 * half;
    v2f fk[8];
    float kn2 = 0.f;
#pragma unroll
    for (int kk = 0; kk < 8; ++kk) {
      v2f t = *(const v2f*)(krow + 4 * kk);
      fk[kk] = t;
      kn2 += t.x * t.x + t.y * t.y;
    }
    kn2 += __shfl_xor(kn2, 16, 32);
    const float kinv = 1.0f / fmaxf(sqrtf(kn2), 1e-12f);
#pragma unroll
    for (int kk = 0; kk < 8; ++kk) fk[kk] *= kinv;

    v8f c = {};
#pragma unroll
    for (int kk = 0; kk < 8; ++kk) c = wmma4(fq[kk], fk[kk], c);

#pragma unroll
    for (int r = 0; r < 8; ++r) {
      const int row = r + 8 * half;          // C layout: lanes>=16 hold M=8..15
      const int m_g = mt * 16 + lr;
      float v = c[r] * sch;
      if (m_g == n0 + row) v = NEGF;          // improve_locality diagonal mask
      sc[row * LDS_STRIDE + m_g] = v;
    }
  }
  __syncthreads();

  // ---- per-row exact top-64 threshold + softmax (wave cooperates) ----
  for (int r = 0; r < 16; ++r) {
    float* srow = sc + r * LDS_STRIDE;
    float vals[33];
    unsigned keys[33];
#pragma unroll
    for (int j = 0; j < 33; ++j) {
      const int m = lane + 32 * j;
      const float v = (m < MTOK) ? srow[m] : NEGF;
      vals[j] = v;
      keys[j] = (m < MTOK) ? f2key(v) : 0u;  // 0 < key(-FLT_MAX): never counted
    }
    float mx = NEGF;
#pragma unroll
    for (int j = 0; j < 33; ++j) mx = fmaxf(mx, vals[j]);
    mx = wred_max(mx);

    // largest key T with count(keys >= T) >= 64  ==> T = key(64th largest)
    unsigned lo = 0u, hi = 0xFFFFFFFFu;
    while (lo < hi) {  // wave-uniform: decision comes from a full reduction
      const unsigned mid = lo + ((hi - lo) >> 1) + 1u;
      int cnt = 0;
#pragma unroll
      for (int j = 0; j < 33; ++j) cnt += (keys[j] >= mid) ? 1 : 0;
      cnt = wred_sumi(cnt);
      if (cnt >= KTOP) lo = mid; else hi = mid - 1u;
    }
    const unsigned T = lo;

    float s = 0.f;
#pragma unroll
    for (int j = 0; j < 33; ++j) {
      const float p = (keys[j] >= T) ? __expf(vals[j] - mx) : 0.f;
      vals[j] = p;   // ties at kth kept, matching `scores < kth -> NEG`
      s += p;
    }
    s = wred_sum(s);
    const float inv = 1.0f / s;
#pragma unroll
    for (int j = 0; j < 33; ++j) {
      const int m = lane + 32 * j;
      if (m < MTOK) srow[m] = vals[j] * inv;
    }
  }
  __syncthreads();

  // ---- out = P (16x1040, LDS) x V (1040x32) via WMMA, two 16-wide d tiles --
  v8f o0 = {}, o1 = {};
  const float* vbaseg = qkv + 2 * SQ + qbase;
  for (int mt = 0; mt < 65; ++mt) {
#pragma unroll
    for (int kk = 0; kk < 4; ++kk) {
      const int mr0 = mt * 16 + 4 * kk + 2 * half;  // this half's two K rows
      v2f a = *(const v2f*)(sc + lr * LDS_STRIDE + mr0);
      const float* v0r = (mt < 64)
                             ? vbaseg + (size_t)mr0 * HD
                             : memv + ((size_t)h * NMEM + (mr0 - NTOK)) * HD;
      const float* v1r = v0r + HD;
      v2f b0, b1;
      b0.x = v0r[lr];      b0.y = v1r[lr];       // d = 0..15
      b1.x = v0r[16 + lr]; b1.y = v1r[16 + lr];  // d = 16..31
      o0 = wmma4(a, b0, o0);
      o1 = wmma4(a, b1, o1);
    }
  }

  // channels-last store: o2[b, n, h*32 + d]
  const size_t obase = ((size_t)b * NTOK) * (HN * HD) + (size_t)h * HD;
#pragma unroll
  for (int r = 0; r < 8; ++r) {
    const int row = r + 8 * half;
    const size_t o = obase + (size_t)(n0 + row) * (HN * HD) + lr;
    o2[o]      = o0[r];
    o2[o + 16] = o1[r];
  }
}

// ---------------------------------------------------------------------------
extern "C" void kernel_launch(void* const* d_in, const int* in_sizes, int n_in,
                              void* d_out, int out_size, void* d_ws,
                              size_t ws_size, hipStream_t stream) {
  (void)in_sizes; (void)n_in; (void)out_size; (void)ws_size;
  const float* x      = (const float*)d_in[0];  // [8,32,32,384]
  const float* w_qkv  = (const float*)d_in[1];  // [1152,384]
  const float* w_proj = (const float*)d_in[2];  // [384,384]
  const float* scale  = (const float*)d_in[3];  // [12]
  const float* memk   = (const float*)d_in[4];  // [12,16,32]
  const float* memv   = (const float*)d_in[5];  // [12,16,32]

  float* qkv = (float*)d_ws;        // 3*SQ floats = 37.75 MB
  float* o2  = qkv + 3 * SQ;        // 8192*384 floats = 12.6 MB

  // qkv = x @ w_qkv^T : [8192,1152]
  gemm_nt<<<dim3(1152 / 64, 8192 / 32), 256, 0, stream>>>(x, w_qkv, qkv, 384,
                                                          1152);
  // fused attention -> channels-last [8192, 384]
  const size_t shmem = (size_t)4 * 16 * LDS_STRIDE * sizeof(float);  // 261 KB
  attn_kernel<<<dim3(8 * HN * (NTOK / 64)), 128, shmem, stream>>>(
      qkv, scale, memk, memv, o2);
  // out = o2 @ w_proj^T : [8192,384]
  gemm_nt<<<dim3(384 / 64, 8192 / 32), 256, 0, stream>>>(o2, w_proj,
                                                         (float*)d_out, 384,
                                                         384);
}